// StructuredAttention_bi_1580547972585
// MI455X (gfx1250) — compile-verified
//
#include <hip/hip_runtime.h>

// ---------------------------------------------------------------------------
// StructuredAttention (bidirectional masked cosine attention), fp32, gfx1250.
// S = Cn @ Qn^T via V_WMMA_F32_16X16X4_F32 (exact fp32 on the matrix pipe),
// row/col masked softmax, then two more fp32 WMMA GEMMs for A_c and A_q.
// Round 2: 32x64 wave tiles (halved VMEM-per-WMMA), parallel col-softmax
// with LDS two-stage reduction and single-pass register caching.
// ---------------------------------------------------------------------------

typedef __attribute__((ext_vector_type(2))) float v2f;
typedef __attribute__((ext_vector_type(8))) float v8f;

constexpr int NI = 16 * 4 * 8;   // B*X*I = 512 independent instances
constexpr int LC = 256;          // context length
constexpr int LQ = 128;          // query/region length
constexpr int HH = 512;          // hidden
constexpr float SCALE_F = 100.0f;
constexpr float NEG_BIG_F = 1.0e10f;

__device__ __forceinline__ float wave_sum(float v) {
#pragma unroll
  for (int off = 16; off > 0; off >>= 1) v += __shfl_xor(v, off, 32);
  return v;
}
__device__ __forceinline__ float wave_max(float v) {
#pragma unroll
  for (int off = 16; off > 0; off >>= 1) v = fmaxf(v, __shfl_xor(v, off, 32));
  return v;
}

// ---------------------------------------------------------------------------
// 1) inverse L2 norms per row (rows of length HH). One wave per row.
// ---------------------------------------------------------------------------
__global__ void rownorm_kernel(const float* __restrict__ X,
                               float* __restrict__ invn) {
  const int lane = threadIdx.x & 31;
  const int row = blockIdx.x * (blockDim.x >> 5) + (threadIdx.x >> 5);
  const float* rp = X + (size_t)row * HH;
  float s = 0.0f;
#pragma unroll
  for (int k = lane * 4; k < HH; k += 128) {
    float4 v = *(const float4*)(rp + k);
    s = fmaf(v.x, v.x, s);
    s = fmaf(v.y, v.y, s);
    s = fmaf(v.z, v.z, s);
    s = fmaf(v.w, v.w, s);
  }
  s = wave_sum(s);
  if (lane == 0) invn[row] = 1.0f / fmaxf(sqrtf(s), 1e-12f);
}

// ---------------------------------------------------------------------------
// 2) logits = (Cn @ Qn^T)*SCALE - SCALE*NEG_BIG*(1-mask).
// Wave computes a 32(M=c) x 64(N=q) tile with fp32 WMMA, K=HH in steps of 4.
// 16 waves per instance (8 M-tiles x 2 N-strips).
// ---------------------------------------------------------------------------
__global__ void score_gemm_kernel(const float* __restrict__ C,
                                  const float* __restrict__ Q,
                                  const float* __restrict__ invc,
                                  const float* __restrict__ invq,
                                  const float* __restrict__ cmask,
                                  const float* __restrict__ qmask,
                                  float* __restrict__ logits) {
  const int lane = threadIdx.x & 31;
  const int gwave = blockIdx.x * (blockDim.x >> 5) + (threadIdx.x >> 5);
  const int inst = gwave >> 4;          // 16 waves per instance
  const int wi = gwave & 15;
  const int m0 = (wi & 7) * 32;         // c tile (32 rows)
  const int n0 = (wi >> 3) * 64;        // q strip
  const float* Ci = C + (size_t)inst * LC * HH;
  const float* Qi = Q + (size_t)inst * LQ * HH;

  v8f acc[2][4];
#pragma unroll
  for (int u = 0; u < 2; ++u)
#pragma unroll
    for (int t = 0; t < 4; ++t) acc[u][t] = v8f{0, 0, 0, 0, 0, 0, 0, 0};

  const int mA = m0 + (lane & 15);
  const int kOff = (lane >> 4) * 2;     // lanes 0-15: K,K+1 ; lanes 16-31: K+2,K+3
  const int nL = lane & 15;

  for (int k = 0; k < HH; k += 4) {
    v2f a0 = *(const v2f*)(Ci + (size_t)mA * HH + k + kOff);
    v2f a1 = *(const v2f*)(Ci + (size_t)(mA + 16) * HH + k + kOff);
#pragma unroll
    for (int t = 0; t < 4; ++t) {
      const int n = n0 + t * 16 + nL;
      // B = Qn^T : B[kk..kk+1][n] == Q[n][kk..kk+1], contiguous in the Q row
      v2f b = *(const v2f*)(Qi + (size_t)n * HH + k + kOff);
      acc[0][t] = __builtin_amdgcn_wmma_f32_16x16x4_f32(
          false, a0, false, b, (short)0, acc[0][t], false, false);
      acc[1][t] = __builtin_amdgcn_wmma_f32_16x16x4_f32(
          false, a1, false, b, (short)0, acc[1][t], false, false);
    }
  }

  // finalize: scale by inverse norms, apply mask penalty, store logits
  float iq[4], qm[4];
#pragma unroll
  for (int t = 0; t < 4; ++t) {
    const int n = n0 + t * 16 + nL;
    iq[t] = invq[(size_t)inst * LQ + n];
    qm[t] = qmask[(size_t)inst * LQ + n];
  }
  const int mHalf = (lane >> 4) * 8;    // C/D layout: lane<16 -> M=j ; lane>=16 -> M=8+j
#pragma unroll
  for (int u = 0; u < 2; ++u) {
#pragma unroll
    for (int j = 0; j < 8; ++j) {
      const int m = m0 + u * 16 + mHalf + j;
      const float ic = invc[(size_t)inst * LC + m];
      const float cm = cmask[(size_t)inst * LC + m];
#pragma unroll
      for (int t = 0; t < 4; ++t) {
        const int n = n0 + t * 16 + nL;
        const float s = acc[u][t][j] * ic * iq[t];
        const float msk = cm * qm[t];
        logits[((size_t)inst * LC + m) * LQ + n] =
            s * SCALE_F - (SCALE_F * NEG_BIG_F) * (1.0f - msk);
      }
    }
  }
}

// ---------------------------------------------------------------------------
// 3) row softmax over q (axis -1) -> S_c ; also emits S_mask output.
// One wave per row of 128.
// ---------------------------------------------------------------------------
__global__ void row_softmax_kernel(const float* __restrict__ logits,
                                   const float* __restrict__ cmask,
                                   const float* __restrict__ qmask,
                                   float* __restrict__ Sc,
                                   float* __restrict__ smask_out) {
  const int lane = threadIdx.x & 31;
  const int row = blockIdx.x * (blockDim.x >> 5) + (threadIdx.x >> 5); // inst*LC + c
  const int inst = row >> 8;
  const int c = row & 255;
  const float* rp = logits + (size_t)row * LQ;

  float v[4];
  float mx = -3.0e38f;
#pragma unroll
  for (int i = 0; i < 4; ++i) {
    v[i] = rp[lane + 32 * i];
    mx = fmaxf(mx, v[i]);
  }
  mx = wave_max(mx);
  float sum = 0.0f;
#pragma unroll
  for (int i = 0; i < 4; ++i) {
    v[i] = __expf(v[i] - mx);
    sum += v[i];
  }
  sum = wave_sum(sum);
  const float inv = 1.0f / sum;
  const float cm = cmask[(size_t)inst * LC + c];
#pragma unroll
  for (int i = 0; i < 4; ++i) {
    const int q = lane + 32 * i;
    const float msk = cm * qmask[(size_t)inst * LQ + q];
    Sc[(size_t)row * LQ + q] = v[i] * inv * msk;
    smask_out[(size_t)row * LQ + q] = msk;
  }
}

// ---------------------------------------------------------------------------
// 4) column softmax over c (axis -2) -> S_q, in place over logits.
// One 1024-thread block per instance: thread (q = t&127, g = t>>7) keeps its
// 32-row slice in registers, two-stage max/sum reduction through LDS,
// single read pass + single write pass. Also emits S_mask^T output.
// ---------------------------------------------------------------------------
__global__ void col_softmax_kernel(float* __restrict__ logits,
                                   const float* __restrict__ cmask,
                                   const float* __restrict__ qmask,
                                   float* __restrict__ smaskT_out) {
  __shared__ float red[8][LQ];
  const int inst = blockIdx.x;
  const int q = threadIdx.x & 127;
  const int g = threadIdx.x >> 7;       // 0..7 row-group
  const int c0 = g * 32;
  float* base = logits + (size_t)inst * LC * LQ + q;

  float vals[32];
  float mx = -3.0e38f;
#pragma unroll
  for (int i = 0; i < 32; ++i) {
    vals[i] = base[(size_t)(c0 + i) * LQ];
    mx = fmaxf(mx, vals[i]);
  }
  red[g][q] = mx;
  __syncthreads();
  float m = red[0][q];
#pragma unroll
  for (int i = 1; i < 8; ++i) m = fmaxf(m, red[i][q]);
  __syncthreads();                      // done reading before reuse

  float sum = 0.0f;
#pragma unroll
  for (int i = 0; i < 32; ++i) {
    vals[i] = __expf(vals[i] - m);
    sum += vals[i];
  }
  red[g][q] = sum;
  __syncthreads();
  float tot = red[0][q];
#pragma unroll
  for (int i = 1; i < 8; ++i) tot += red[i][q];

  const float inv = 1.0f / tot;
  const float qm = qmask[(size_t)inst * LQ + q];
#pragma unroll
  for (int i = 0; i < 32; ++i) {
    const int c = c0 + i;
    const float msk = qm * cmask[(size_t)inst * LC + c];
    base[(size_t)c * LQ] = vals[i] * inv * msk;
    smaskT_out[((size_t)inst * LQ + q) * LC + c] = msk;
  }
}

// ---------------------------------------------------------------------------
// 5a) A_c = S_c @ Q : M=LC, K=LQ, N=HH. Wave does 32x64. 64 waves/instance.
// ---------------------------------------------------------------------------
__global__ void ac_gemm_kernel(const float* __restrict__ Sc,
                               const float* __restrict__ Q,
                               float* __restrict__ Ac) {
  const int lane = threadIdx.x & 31;
  const int gwave = blockIdx.x * (blockDim.x >> 5) + (threadIdx.x >> 5);
  const int inst = gwave >> 6;          // 64 waves per instance
  const int wi = gwave & 63;
  const int m0 = (wi & 7) * 32;         // c tile (32 rows)
  const int n0 = (wi >> 3) * 64;        // h strip
  const float* Ai = Sc + (size_t)inst * LC * LQ;
  const float* Bi = Q + (size_t)inst * LQ * HH;

  v8f acc[2][4];
#pragma unroll
  for (int u = 0; u < 2; ++u)
#pragma unroll
    for (int t = 0; t < 4; ++t) acc[u][t] = v8f{0, 0, 0, 0, 0, 0, 0, 0};

  const int mA = m0 + (lane & 15);
  const int kOff = (lane >> 4) * 2;
  const int nL = lane & 15;

  for (int k = 0; k < LQ; k += 4) {
    const int kk = k + kOff;
    v2f a0 = *(const v2f*)(Ai + (size_t)mA * LQ + kk);
    v2f a1 = *(const v2f*)(Ai + (size_t)(mA + 16) * LQ + kk);
#pragma unroll
    for (int t = 0; t < 4; ++t) {
      const int n = n0 + t * 16 + nL;
      v2f b;
      b.x = Bi[(size_t)kk * HH + n];          // B row-major: stride HH between K rows
      b.y = Bi[(size_t)(kk + 1) * HH + n];
      acc[0][t] = __builtin_amdgcn_wmma_f32_16x16x4_f32(
          false, a0, false, b, (short)0, acc[0][t], false, false);
      acc[1][t] = __builtin_amdgcn_wmma_f32_16x16x4_f32(
          false, a1, false, b, (short)0, acc[1][t], false, false);
    }
  }

  const int mHalf = (lane >> 4) * 8;
#pragma unroll
  for (int u = 0; u < 2; ++u)
#pragma unroll
    for (int j = 0; j < 8; ++j) {
      const int m = m0 + u * 16 + mHalf + j;
#pragma unroll
      for (int t = 0; t < 4; ++t) {
        const int n = n0 + t * 16 + nL;
        Ac[((size_t)inst * LC + m) * HH + n] = acc[u][t][j];
      }
    }
}

// ---------------------------------------------------------------------------
// 5b) A_q = S_q^T @ C : M=LQ, K=LC, N=HH. Wave does 32x64. 32 waves/instance.
// A[m][k] = S_q[k][m] (strided reads from the in-place S_q buffer).
// ---------------------------------------------------------------------------
__global__ void aq_gemm_kernel(const float* __restrict__ Sq,
                               const float* __restrict__ C,
                               float* __restrict__ Aq) {
  const int lane = threadIdx.x & 31;
  const int gwave = blockIdx.x * (blockDim.x >> 5) + (threadIdx.x >> 5);
  const int inst = gwave >> 5;          // 32 waves per instance
  const int wi = gwave & 31;
  const int m0 = (wi & 3) * 32;         // q tile (LQ/32 = 4)
  const int n0 = (wi >> 2) * 64;        // h strip
  const float* Si = Sq + (size_t)inst * LC * LQ;
  const float* Bi = C + (size_t)inst * LC * HH;

  v8f acc[2][4];
#pragma unroll
  for (int u = 0; u < 2; ++u)
#pragma unroll
    for (int t = 0; t < 4; ++t) acc[u][t] = v8f{0, 0, 0, 0, 0, 0, 0, 0};

  const int mA = m0 + (lane & 15);
  const int kOff = (lane >> 4) * 2;
  const int nL = lane & 15;

  for (int k = 0; k < LC; k += 4) {
    const int kk = k + kOff;
    v2f a0, a1;
    a0.x = Si[(size_t)kk * LQ + mA];          // S_q^T[m][k] = S_q[k][m]
    a0.y = Si[(size_t)(kk + 1) * LQ + mA];
    a1.x = Si[(size_t)kk * LQ + mA + 16];
    a1.y = Si[(size_t)(kk + 1) * LQ + mA + 16];
#pragma unroll
    for (int t = 0; t < 4; ++t) {
      const int n = n0 + t * 16 + nL;
      v2f b;
      b.x = Bi[(size_t)kk * HH + n];
      b.y = Bi[(size_t)(kk + 1) * HH + n];
      acc[0][t] = __builtin_amdgcn_wmma_f32_16x16x4_f32(
          false, a0, false, b, (short)0, acc[0][t], false, false);
      acc[1][t] = __builtin_amdgcn_wmma_f32_16x16x4_f32(
          false, a1, false, b, (short)0, acc[1][t], false, false);
    }
  }

  const int mHalf = (lane >> 4) * 8;
#pragma unroll
  for (int u = 0; u < 2; ++u)
#pragma unroll
    for (int j = 0; j < 8; ++j) {
      const int m = m0 + u * 16 + mHalf + j;
#pragma unroll
      for (int t = 0; t < 4; ++t) {
        const int n = n0 + t * 16 + nL;
        Aq[((size_t)inst * LQ + m) * HH + n] = acc[u][t][j];
      }
    }
}

// ---------------------------------------------------------------------------
// launch
// ---------------------------------------------------------------------------
extern "C" void kernel_launch(void* const* d_in, const int* in_sizes, int n_in,
                              void* d_out, int out_size, void* d_ws, size_t ws_size,
                              hipStream_t stream) {
  const float* C = (const float*)d_in[0];      // [NI, LC, HH]
  const float* Q = (const float*)d_in[1];      // [NI, LQ, HH]
  const float* cmask = (const float*)d_in[2];  // [NI, LC]
  const float* qmask = (const float*)d_in[3];  // [NI, LQ]

  float* out = (float*)d_out;
  float* Ac = out;                                      // NI*LC*HH
  float* Aq = Ac + (size_t)NI * LC * HH;                // NI*LQ*HH
  float* Smask = Aq + (size_t)NI * LQ * HH;             // NI*LC*LQ
  float* SmaskT = Smask + (size_t)NI * LC * LQ;         // NI*LQ*LC

  // workspace: logits->S_q (64MB), S_c (64MB), invc (512KB), invq (256KB)
  float* ws = (float*)d_ws;
  float* logits = ws;                                   // NI*LC*LQ
  float* Sc = logits + (size_t)NI * LC * LQ;            // NI*LC*LQ
  float* invc = Sc + (size_t)NI * LC * LQ;              // NI*LC
  float* invq = invc + (size_t)NI * LC;                 // NI*LQ

  // 1) inverse row norms (wave per row, 8 waves per 256-thread block)
  rownorm_kernel<<<(NI * LC) / 8, 256, 0, stream>>>(C, invc);
  rownorm_kernel<<<(NI * LQ) / 8, 256, 0, stream>>>(Q, invq);

  // 2) masked/scaled logits via fp32 WMMA (16 waves/inst, 4 waves/block)
  score_gemm_kernel<<<(NI * 16) / 4, 128, 0, stream>>>(C, Q, invc, invq, cmask,
                                                       qmask, logits);

  // 3) row softmax -> S_c (+ S_mask)
  row_softmax_kernel<<<(NI * LC) / 8, 256, 0, stream>>>(logits, cmask, qmask,
                                                        Sc, Smask);

  // 4) column softmax -> S_q in place (+ S_mask^T); runs after (3) in-stream
  col_softmax_kernel<<<NI, 1024, 0, stream>>>(logits, cmask, qmask, SmaskT);

  // 5) output GEMMs via fp32 WMMA (32x64 wave tiles)
  ac_gemm_kernel<<<(NI * 64) / 4, 128, 0, stream>>>(Sc, Q, Ac);
  aq_gemm_kernel<<<(NI * 32) / 4, 128, 0, stream>>>(logits, C, Aq);
}